// CMAESAlignment_56831007261079
// MI455X (gfx1250) — compile-verified
//
#include <hip/hip_runtime.h>
#include <hip/hip_bf16.h>
#include <math.h>

#define B_    32
#define N_    16384
#define S_    256
#define P_    50
#define MU_   25
#define ITERS_ 30
#define ROT_  0.7853981633974483f
#define TR_   1.0f
#define CSIG_ 0.3f
#define DSIG_ 1.0f
#define BIGF_ 3.402823e38f
#define WSUM_ 22.9633560f  // sum_{i=1..25} (ln(25.5) - ln(i))

typedef __attribute__((ext_vector_type(2))) float v2f;
typedef __attribute__((ext_vector_type(8))) float v8f;

// ---------- helpers ----------
__device__ __forceinline__ unsigned pcg_hash(unsigned x) {
    x = x * 747796405u + 2891336453u;
    unsigned w = ((x >> ((x >> 28u) + 4u)) ^ x) * 277803737u;
    return (w >> 22u) ^ w;
}

__device__ __forceinline__ void aa_to_R(const float a0, const float a1, const float a2, float R[9]) {
    float th = sqrtf(a0 * a0 + a1 * a1 + a2 * a2);
    float inv = 1.0f / (th + 1e-8f);
    float kx = a0 * inv, ky = a1 * inv, kz = a2 * inv;
    float s = sinf(th), c = cosf(th), omc = 1.0f - c;
    float kk = kx * kx + ky * ky + kz * kz;
    R[0] = 1.0f + omc * (kx * kx - kk); R[1] = -s * kz + omc * kx * ky;     R[2] = s * ky + omc * kx * kz;
    R[3] = s * kz + omc * ky * kx;      R[4] = 1.0f + omc * (ky * ky - kk); R[5] = -s * kx + omc * ky * kz;
    R[6] = -s * ky + omc * kz * kx;     R[7] = s * kx + omc * kz * ky;      R[8] = 1.0f + omc * (kz * kz - kk);
}

// ---------- init: subsample + CMA state ----------
__global__ __launch_bounds__(256) void init_k(const float* __restrict__ src, const float* __restrict__ tgt,
                                              float* src_sub, float* tgt_sub, float* mean, float* sigma,
                                              float* gpos, float* gfit) {
    int b = blockIdx.x, j = threadIdx.x;
    int idx = j * (N_ / S_);
    const float* sp = src + ((size_t)b * N_ + idx) * 3;
    ((float4*)src_sub)[b * S_ + j] = make_float4(sp[0], sp[1], sp[2], 0.0f);
    const float* tp = tgt + ((size_t)b * N_ + idx) * 3;
    float tx = tp[0], ty = tp[1], tz = tp[2];
    ((float4*)tgt_sub)[b * S_ + j] = make_float4(tx, ty, tz, tx * tx + ty * ty + tz * tz);
    if (j < 6) {
        mean[b * 6 + j] = 0.0f;
        sigma[b * 6 + j] = (j < 3) ? ROT_ * 0.3f : TR_ * 0.3f;
        gpos[b * 6 + j] = 0.0f;
    }
    if (j == 0) gfit[b] = BIGF_;
}

// ---------- sample population ----------
__global__ void sample_k(const float* __restrict__ mean, const float* __restrict__ sigma,
                         float* __restrict__ z, float* __restrict__ off, int iter) {
    int b = blockIdx.x, t = threadIdx.x;
    if (t >= P_ * 6) return;
    int p = t / 6, d = t % 6;
    unsigned s = (unsigned)(((b * P_ + p) * 6 + d) * 9781 + iter * 7919 + 12345);
    unsigned h1 = pcg_hash(s);
    unsigned h2 = pcg_hash(s ^ 0x9e3779b9u);
    float u1 = ((float)h1 + 1.0f) * 2.3283064365386963e-10f;
    float u2 = (float)h2 * 2.3283064365386963e-10f;
    float zn = sqrtf(-2.0f * logf(u1)) * cosf(6.283185307179586f * u2);
    float v = mean[b * 6 + d] + sigma[b * 6 + d] * zn;
    float lim = (d < 3) ? ROT_ : TR_;
    v = fminf(fmaxf(v, -lim), lim);
    z[(b * P_ + p) * 6 + d] = zn;
    off[(b * P_ + p) * 6 + d] = v;
}

// ---------- chamfer evaluation via V_WMMA_F32_16X16X4_F32 ----------
// d2(M,N) = |x_M|^2 + |y_N|^2 - 2 x_M.y_N computed as one WMMA per 16x16 tile:
//   A row = (-2x0,-2x1,-2x2, 1), B col = (y0,y1,y2,|y|^2), C init = |x|^2.
// Wave layout (wave32): lanes 0-15 hold K-pair {0,1}, lanes 16-31 hold K-pair {2,3}.
__global__ __launch_bounds__(256) void eval_k(const float* __restrict__ srcs, const float* __restrict__ tgts,
                                              const float* __restrict__ posi, float* __restrict__ fit, int nP) {
    int blk = blockIdx.x;
    int b = blk / nP, p = blk % nP;
    int tid = threadIdx.x;

    __shared__ float4 xs[S_];
    __shared__ float4 ys[S_];
    __shared__ unsigned cmin[S_];
    __shared__ float rmn[S_];
    __shared__ float wpart[8];

    const float* g = posi + (b * nP + p) * 6;
    float R[9];
    aa_to_R(g[0], g[1], g[2], R);
    float t0 = g[3], t1 = g[4], t2 = g[5];

    const float* sp = srcs + ((size_t)b * S_ + tid) * 4;
    float sx = sp[0], sy = sp[1], sz = sp[2];
    float x0 = R[0] * sx + R[1] * sy + R[2] * sz + t0;
    float x1 = R[3] * sx + R[4] * sy + R[5] * sz + t1;
    float x2 = R[6] * sx + R[7] * sy + R[8] * sz + t2;
    xs[tid] = make_float4(x0, x1, x2, x0 * x0 + x1 * x1 + x2 * x2);
    const float* tp = tgts + ((size_t)b * S_ + tid) * 4;
    ys[tid] = make_float4(tp[0], tp[1], tp[2], tp[3]);
    cmin[tid] = 0x7f800000u;  // +inf bits; d2 >= 0 so uint-min == float-min
    __syncthreads();

    int lane = tid & 31, wave = tid >> 5;
    int half = lane >> 4, lr = lane & 15;

    #pragma unroll
    for (int mi = 0; mi < 2; ++mi) {
        int mt = wave * 2 + mi;
        // A fragment: contiguous K pair from xs row, patched for K=3 -> 1.0
        v2f araw = *(const v2f*)((const float*)&xs[mt * 16 + lr] + half * 2);
        v2f A;
        A.x = -2.0f * araw.x;
        A.y = half ? 1.0f : -2.0f * araw.y;
        float c8[8];
        #pragma unroll
        for (int r = 0; r < 8; ++r) c8[r] = xs[mt * 16 + half * 8 + r].w;  // |x_M|^2
        float rm[8];
        #pragma unroll
        for (int r = 0; r < 8; ++r) rm[r] = BIGF_;

        for (int nt = 0; nt < 16; ++nt) {
            // B fragment is literally the contiguous K pair of ys (K=3 slot holds |y|^2)
            v2f Bv = *(const v2f*)((const float*)&ys[nt * 16 + lr] + half * 2);
            v8f c;
            #pragma unroll
            for (int r = 0; r < 8; ++r) c[r] = c8[r];
            v8f d = __builtin_amdgcn_wmma_f32_16x16x4_f32(false, A, false, Bv,
                                                          (short)0, c, false, false);
            float cm = d[0];
            #pragma unroll
            for (int r = 1; r < 8; ++r) cm = fminf(cm, d[r]);
            #pragma unroll
            for (int r = 0; r < 8; ++r) rm[r] = fminf(rm[r], d[r]);
            // both halves contribute their 8-row partial min; LDS atomic merges them.
            // no shuffle, no divergence: EXEC stays all-1s for the whole WMMA loop.
            atomicMin(&cmin[nt * 16 + lr], __float_as_uint(cm));
        }
        // row-min: reduce across the 16 lanes of each half
        #pragma unroll
        for (int r = 0; r < 8; ++r) {
            rm[r] = fminf(rm[r], __shfl_xor(rm[r], 8, 32));
            rm[r] = fminf(rm[r], __shfl_xor(rm[r], 4, 32));
            rm[r] = fminf(rm[r], __shfl_xor(rm[r], 2, 32));
            rm[r] = fminf(rm[r], __shfl_xor(rm[r], 1, 32));
        }
        if (lr == 0) {
            #pragma unroll
            for (int r = 0; r < 8; ++r) rmn[mt * 16 + half * 8 + r] = rm[r];
        }
    }
    __syncthreads();

    float v = rmn[tid] * (1.0f / S_) + __uint_as_float(cmin[tid]) * (1.0f / S_);
    v += __shfl_xor(v, 16, 32);
    v += __shfl_xor(v, 8, 32);
    v += __shfl_xor(v, 4, 32);
    v += __shfl_xor(v, 2, 32);
    v += __shfl_xor(v, 1, 32);
    if (lane == 0) wpart[wave] = v;
    __syncthreads();
    if (tid == 0) {
        float s = 0.0f;
        for (int i = 0; i < 8; ++i) s += wpart[i];
        fit[b * nP + p] = s;
    }
}

// ---------- CMA-ES update: rank-select, recombine, sigma, gbest ----------
__global__ void update_k(const float* __restrict__ fit, const float* __restrict__ z,
                         const float* __restrict__ off, float* mean, float* sigma,
                         float* gpos, float* gfit) {
    int b = blockIdx.x, t = threadIdx.x;
    __shared__ float sfit[P_];
    __shared__ float accp[6], accz[6];
    __shared__ int besti;
    if (t < P_) sfit[t] = fit[b * P_ + t];
    if (t < 6) { accp[t] = 0.0f; accz[t] = 0.0f; }
    __syncthreads();
    if (t < P_) {
        float f = sfit[t];
        int rank = 0;
        for (int j = 0; j < P_; ++j) {
            float fj = sfit[j];
            rank += (fj < f) || (fj == f && j < t);
        }
        if (rank == 0) besti = t;
        if (rank < MU_) {
            float w = (logf((float)MU_ + 0.5f) - logf((float)(rank + 1))) * (1.0f / WSUM_);
            for (int d = 0; d < 6; ++d) {
                atomicAdd(&accp[d], w * off[(b * P_ + t) * 6 + d]);
                atomicAdd(&accz[d], w * z[(b * P_ + t) * 6 + d]);
            }
        }
    }
    __syncthreads();
    if (t == 0) {
        float mz = 0.0f;
        for (int d = 0; d < 6; ++d) mz += fabsf(accz[d]);
        mz *= (1.0f / 6.0f);
        float sc = expf((CSIG_ / DSIG_) * (mz - 1.0f));
        for (int d = 0; d < 6; ++d) {
            mean[b * 6 + d] = accp[d];
            float s = sigma[b * 6 + d] * sc;
            float lo = (d < 3) ? ROT_ * 0.01f : TR_ * 0.01f;
            float hi = (d < 3) ? ROT_ * 0.5f : TR_ * 0.5f;
            sigma[b * 6 + d] = fminf(fmaxf(s, lo), hi);
        }
        float bf = sfit[besti];
        if (bf < gfit[b]) {
            gfit[b] = bf;
            for (int d = 0; d < 6; ++d) gpos[b * 6 + d] = off[(b * P_ + besti) * 6 + d];
        }
    }
}

// ---------- final pose selection, write R and t ----------
__global__ void pick_k(const float* __restrict__ mean, const float* __restrict__ mfit,
                       const float* __restrict__ gfit, const float* __restrict__ gpos,
                       float* pose, float* out) {
    int b = blockIdx.x;
    if (threadIdx.x != 0) return;
    const float* g = (mfit[b] < gfit[b]) ? (mean + b * 6) : (gpos + b * 6);
    float R[9];
    aa_to_R(g[0], g[1], g[2], R);
    for (int i = 0; i < 9; ++i) { pose[b * 12 + i] = R[i]; out[b * 9 + i] = R[i]; }
    for (int j = 0; j < 3; ++j) { pose[b * 12 + 9 + j] = g[3 + j]; out[B_ * 9 + b * 3 + j] = g[3 + j]; }
}

// ---------- apply final transform to full cloud ----------
__global__ __launch_bounds__(256) void transform_k(const float* __restrict__ src,
                                                   const float* __restrict__ pose,
                                                   float* __restrict__ out) {
    size_t idx = (size_t)blockIdx.x * 256 + threadIdx.x;  // over B_*N_
    int b = (int)(idx >> 14);
    int n = (int)(idx & (N_ - 1));
    const float* Rp = pose + b * 12;
    const float* s = src + ((size_t)b * N_ + n) * 3;
    float sx = s[0], sy = s[1], sz = s[2];
    float* o = out + B_ * 9 + B_ * 3 + ((size_t)b * N_ + n) * 3;
    o[0] = Rp[0] * sx + Rp[1] * sy + Rp[2] * sz + Rp[9];
    o[1] = Rp[3] * sx + Rp[4] * sy + Rp[5] * sz + Rp[10];
    o[2] = Rp[6] * sx + Rp[7] * sy + Rp[8] * sz + Rp[11];
}

extern "C" void kernel_launch(void* const* d_in, const int* in_sizes, int n_in,
                              void* d_out, int out_size, void* d_ws, size_t ws_size,
                              hipStream_t stream) {
    const float* source = (const float*)d_in[0];
    const float* target = (const float*)d_in[1];
    float* out = (float*)d_out;

    float* w = (float*)d_ws;
    float* src_sub = w;                     // B_*S_*4 = 32768
    float* tgt_sub = src_sub + B_ * S_ * 4; // 32768
    float* mean    = tgt_sub + B_ * S_ * 4; // 192
    float* sigma   = mean + B_ * 6;         // 192
    float* gpos    = sigma + B_ * 6;        // 192
    float* gfit    = gpos + B_ * 6;         // 32
    float* zbuf    = gfit + B_;             // 9600
    float* offbuf  = zbuf + B_ * P_ * 6;    // 9600
    float* fitbuf  = offbuf + B_ * P_ * 6;  // 1600
    float* mfit    = fitbuf + B_ * P_;      // 32
    float* pose    = mfit + B_;             // 384

    init_k<<<B_, S_, 0, stream>>>(source, target, src_sub, tgt_sub, mean, sigma, gpos, gfit);

    for (int it = 0; it < ITERS_; ++it) {
        sample_k<<<B_, 320, 0, stream>>>(mean, sigma, zbuf, offbuf, it);
        eval_k<<<B_ * P_, 256, 0, stream>>>(src_sub, tgt_sub, offbuf, fitbuf, P_);
        update_k<<<B_, 64, 0, stream>>>(fitbuf, zbuf, offbuf, mean, sigma, gpos, gfit);
    }

    eval_k<<<B_, 256, 0, stream>>>(src_sub, tgt_sub, mean, mfit, 1);
    pick_k<<<B_, 32, 0, stream>>>(mean, mfit, gfit, gpos, pose, out);
    transform_k<<<(B_ * N_) / 256, 256, 0, stream>>>(source, pose, out);
}